// GAT_2491081032171
// MI455X (gfx1250) — compile-verified
//
#include <hip/hip_runtime.h>
#include <hip/hip_bf16.h>

typedef __attribute__((ext_vector_type(2))) float v2f;
typedef __attribute__((ext_vector_type(8))) float v8f;

#define NEG_SLOPE 0.2f

static __device__ __forceinline__ int imin(int a, int b) { return a < b ? a : b; }

// Order-preserving float <-> unsigned encoding for atomicMax on float data.
static __device__ __forceinline__ unsigned fenc(float f) {
    unsigned b = __float_as_uint(f);
    return (b & 0x80000000u) ? ~b : (b | 0x80000000u);
}
static __device__ __forceinline__ float fdec(unsigned u) {
    unsigned b = (u & 0x80000000u) ? (u & 0x7fffffffu) : ~u;
    return __uint_as_float(b);
}

// ---------------------------------------------------------------------------
// GEMM: Hout[N, NOUT] = A[N, K] @ W[K, NOUT] using V_WMMA_F32_16X16X4_F32.
// Block = 256 threads (8 wave32). Each wave computes 16 rows x NCT*16 cols.
// W^T is staged in LDS in K-chunks of 128 so B-operand pairs are contiguous.
// K must be a multiple of KC = 128 (no K tail is handled)
// (here K in {256,128}).
// ---------------------------------------------------------------------------
template <int NCT>
__global__ __launch_bounds__(256)
void gemm_wmma_kernel(const float* __restrict__ A, const float* __restrict__ W,
                      float* __restrict__ Hout, int N, int K, int NOUT)
{
    constexpr int KC = 128;
    constexpr int NOUT_PAD = NCT * 16;
    __shared__ float ldsWT[NOUT_PAD * KC];   // NCT=8 -> 64KB, NCT=3 -> 24KB

    const int tid  = threadIdx.x;
    const int wave = tid >> 5;
    const int lane = tid & 31;
    const int l16  = lane & 15;
    const int half = lane >> 4;          // 0: K pair (k0,k0+1), 1: (k0+2,k0+3)

    const int rowTile = blockIdx.x * 128 + wave * 16;
    const int rowA    = imin(rowTile + l16, N - 1);   // clamp: EXEC stays all-1 for WMMA

    v8f acc[NCT];
    v8f vz = {0.f, 0.f, 0.f, 0.f, 0.f, 0.f, 0.f, 0.f};
#pragma unroll
    for (int ct = 0; ct < NCT; ++ct) acc[ct] = vz;

    for (int kc = 0; kc < K; kc += KC) {
        // Stage W^T chunk: ldsWT[n*KC + k] = W[(kc+k)*NOUT + n], zero-pad n>=NOUT.
        for (int i = tid; i < NOUT_PAD * KC; i += 256) {
            int n = i % NOUT_PAD;       // consecutive lanes -> consecutive n (coalesced)
            int k = i / NOUT_PAD;
            float w = (n < NOUT) ? W[(size_t)(kc + k) * NOUT + n] : 0.0f;
            ldsWT[n * KC + k] = w;
        }
        __syncthreads();

        const float* Arow = A + (size_t)rowA * K + kc;
        // speculative prefetch of next A chunk (global_prefetch_b8)
        __builtin_prefetch(Arow + KC, 0, 0);

#pragma unroll 4
        for (int k0 = 0; k0 < KC; k0 += 4) {
            const int kb = k0 + half * 2;
            // A 16x4 f32 fragment: lane<16 holds (K=k0,k0+1); lane>=16 (k0+2,k0+3)
            v2f av = *(const v2f*)(Arow + kb);
#pragma unroll
            for (int ct = 0; ct < NCT; ++ct) {
                v2f bv = *(const v2f*)(&ldsWT[(ct * 16 + l16) * KC + kb]);
                acc[ct] = __builtin_amdgcn_wmma_f32_16x16x4_f32(
                    false, av, false, bv, (short)0, acc[ct], false, false);
            }
        }
        __syncthreads();
    }

    // C/D layout: VGPR g -> row = rowTile + g + 8*half, col = ct*16 + l16
    const int rstore = rowTile + half * 8;
#pragma unroll
    for (int ct = 0; ct < NCT; ++ct) {
        const int c = ct * 16 + l16;
        if (c >= NOUT) continue;
#pragma unroll
        for (int g = 0; g < 8; ++g) {
            const int r = rstore + g;
            if (r < N) Hout[(size_t)r * NOUT + c] = acc[ct][g];
        }
    }
}

// ---------------------------------------------------------------------------
// el[n,h] = sum_d h[n,h,d]*al[h,d] ; er likewise
// ---------------------------------------------------------------------------
__global__ void eler_kernel(const float* __restrict__ Hfeat,
                            const float* __restrict__ al, const float* __restrict__ ar,
                            float* __restrict__ el, float* __restrict__ er,
                            int N, int H, int D)
{
    long long idx = (long long)blockIdx.x * blockDim.x + threadIdx.x;
    if (idx >= (long long)N * H) return;
    int n = (int)(idx / H), h = (int)(idx % H);
    const float* hp = Hfeat + (size_t)n * H * D + (size_t)h * D;
    float sl = 0.f, sr = 0.f;
    for (int d = 0; d < D; ++d) {
        float v = hp[d];
        sl += v * al[h * D + d];
        sr += v * ar[h * D + d];
    }
    el[idx] = sl;
    er[idx] = sr;
}

// e = leaky_relu(el[src]+er[dst]); ebuf = e; emax[dst] = max(emax[dst], e)
__global__ void edge_max_kernel(const int* __restrict__ src, const int* __restrict__ dst,
                                const float* __restrict__ el, const float* __restrict__ er,
                                float* __restrict__ ebuf, unsigned* __restrict__ emax,
                                long long E, int H)
{
    long long idx = (long long)blockIdx.x * blockDim.x + threadIdx.x;
    if (idx >= E * H) return;
    long long e_id = idx / H;
    int h = (int)(idx % H);
    int s = src[e_id], d = dst[e_id];
    float e = el[(size_t)s * H + h] + er[(size_t)d * H + h];
    e = (e > 0.f) ? e : NEG_SLOPE * e;
    ebuf[idx] = e;
    atomicMax(&emax[(size_t)d * H + h], fenc(e));
}

// ee = exp(e - emax[dst]); ebuf = ee; denom[dst] += ee
__global__ void edge_expsum_kernel(const int* __restrict__ dst,
                                   float* __restrict__ ebuf,
                                   const unsigned* __restrict__ emax,
                                   float* __restrict__ denom,
                                   long long E, int H)
{
    long long idx = (long long)blockIdx.x * blockDim.x + threadIdx.x;
    if (idx >= E * H) return;
    long long e_id = idx / H;
    int h = (int)(idx % H);
    int d = dst[e_id];
    float m = fdec(emax[(size_t)d * H + h]);
    float ee = expf(ebuf[idx] - m);
    ebuf[idx] = ee;
    atomicAdd(&denom[(size_t)d * H + h], ee);
}

// O[dst, f] += h[src, f] * (ee / max(denom[dst,h], 1e-9))
__global__ void edge_scatter_kernel(const int* __restrict__ src, const int* __restrict__ dst,
                                    const float* __restrict__ Hfeat,
                                    const float* __restrict__ ebuf,
                                    const float* __restrict__ denom,
                                    float* __restrict__ O,
                                    long long E, int H, int D)
{
    const int F = H * D;
    long long idx = (long long)blockIdx.x * blockDim.x + threadIdx.x;
    if (idx >= E * F) return;
    long long e_id = idx / F;
    int f = (int)(idx % F);
    int h = f / D;
    int s = src[e_id], d = dst[e_id];
    float alpha = ebuf[e_id * H + h] / fmaxf(denom[(size_t)d * H + h], 1e-9f);
    atomicAdd(&O[(size_t)d * F + f], Hfeat[(size_t)s * F + f] * alpha);
}

__global__ void elu_kernel(float* __restrict__ O, long long n)
{
    long long idx = (long long)blockIdx.x * blockDim.x + threadIdx.x;
    if (idx >= n) return;
    float x = O[idx];
    O[idx] = (x > 0.f) ? x : (expf(x) - 1.0f);
}

// ---------------------------------------------------------------------------
static void run_layer(const float* X, int K, const float* W,
                      const float* al, const float* ar, int H, int D,
                      float* O, bool do_elu,
                      const int* src, const int* dst, int N, long long E,
                      float* h_buf, float* el, float* er,
                      unsigned* emax, float* denom, float* ebuf,
                      hipStream_t stream)
{
    const int F = H * D;
    hipMemsetAsync(O, 0, (size_t)N * F * sizeof(float), stream);
    hipMemsetAsync(emax, 0, (size_t)N * H * sizeof(unsigned), stream);
    hipMemsetAsync(denom, 0, (size_t)N * H * sizeof(float), stream);

    const int gblocks = (N + 127) / 128;
    if (F == 128)
        gemm_wmma_kernel<8><<<gblocks, 256, 0, stream>>>(X, W, h_buf, N, K, F);
    else
        gemm_wmma_kernel<3><<<gblocks, 256, 0, stream>>>(X, W, h_buf, N, K, F);

    long long nh = (long long)N * H;
    eler_kernel<<<(unsigned)((nh + 255) / 256), 256, 0, stream>>>(h_buf, al, ar, el, er, N, H, D);

    long long eh = E * H;
    edge_max_kernel<<<(unsigned)((eh + 255) / 256), 256, 0, stream>>>(src, dst, el, er, ebuf, emax, E, H);
    edge_expsum_kernel<<<(unsigned)((eh + 255) / 256), 256, 0, stream>>>(dst, ebuf, emax, denom, E, H);

    long long ef = E * F;
    edge_scatter_kernel<<<(unsigned)((ef + 255) / 256), 256, 0, stream>>>(src, dst, h_buf, ebuf, denom, O, E, H, D);

    if (do_elu) {
        long long nf = (long long)N * F;
        elu_kernel<<<(unsigned)((nf + 255) / 256), 256, 0, stream>>>(O, nf);
    }
}

extern "C" void kernel_launch(void* const* d_in, const int* in_sizes, int n_in,
                              void* d_out, int out_size, void* d_ws, size_t ws_size,
                              hipStream_t stream)
{
    (void)n_in; (void)out_size; (void)ws_size;

    const float* feats = (const float*)d_in[0];
    const int*   src   = (const int*)d_in[1];
    const int*   dst   = (const int*)d_in[2];
    const float* W0    = (const float*)d_in[3];
    const float* al0   = (const float*)d_in[4];
    const float* ar0   = (const float*)d_in[5];
    const float* W1    = (const float*)d_in[6];
    const float* al1   = (const float*)d_in[7];
    const float* ar1   = (const float*)d_in[8];
    const float* W2    = (const float*)d_in[9];
    const float* al2   = (const float*)d_in[10];
    const float* ar2   = (const float*)d_in[11];

    const int N = in_sizes[0] / 256;
    const long long E = in_sizes[1];

    // d_out layout (reference return order): logits[N,40], h1[N,128], h2[N,128]
    float* logits = (float*)d_out;
    float* h1 = logits + (size_t)N * 40;
    float* h2 = h1 + (size_t)N * 128;

    // workspace carve-up
    float*    h_buf = (float*)d_ws;                         // N*128
    float*    el    = h_buf + (size_t)N * 128;              // N*4
    float*    er    = el + (size_t)N * 4;                   // N*4
    unsigned* emax  = (unsigned*)(er + (size_t)N * 4);      // N*4
    float*    denom = (float*)(emax + (size_t)N * 4);       // N*4
    float*    ebuf  = denom + (size_t)N * 4;                // E*4

    // Layer 0: feats[K=256] -> h1[N,128], heads=4, D=32, ELU
    run_layer(feats, 256, W0, al0, ar0, 4, 32, h1, true,
              src, dst, N, E, h_buf, el, er, emax, denom, ebuf, stream);
    // Layer 1: h1[K=128] -> h2[N,128], heads=4, D=32, ELU
    run_layer(h1, 128, W1, al1, ar1, 4, 32, h2, true,
              src, dst, N, E, h_buf, el, er, emax, denom, ebuf, stream);
    // Layer 2: h2[K=128] -> logits[N,40], heads=1, D=40, no activation (mean over 1 head = id)
    run_layer(h2, 128, W2, al2, ar2, 1, 40, logits, false,
              src, dst, N, E, h_buf, el, er, emax, denom, ebuf, stream);
}